// QuantumPhysicsEngine_57939108823231
// MI455X (gfx1250) — compile-verified
//
#include <hip/hip_runtime.h>
#include <math.h>

// ---------------------------------------------------------------------------
// QuantumPhysicsEngine for MI455X (gfx1250, wave32).
//
// psi_k(x)   = sum_j tanh(x*w1_j + b1_j) * W2[j,k] + b2_k
// psi''_k(x) = sum_j [-2 h_j (1-h_j^2)] * w1_j^2 * W2[j,k]
//
// Both are length-256 f32 contractions -> V_WMMA_F32_16X16X4_F32 (the native
// f32 WMMA shape). One wave owns a 16-point tile (M=16); K is chunked by 4
// hidden units. Two WMMAs per chunk accumulate into one 16x16 f32 tile:
//   cols 0,1 <- A=h, B=W2          (psi contraction)
//   cols 2,3 <- A=u, B=w1^2 * W2   (laplacian contraction)
//
// A-matrix 16x4 f32 layout (ISA 7.12.2): lane = M + 16*(K>=2), VGPR v = K&1.
// B-matrix 4x16 assumed mirrored:        lane = N + 16*(K>=2), VGPR v = K&1.
// C/D 16x16 f32: VGPR r, lanes 0-15 -> M=r, N=lane; lanes 16-31 -> M=r+8.
//
// Round-1 fix: B-element selection is done with per-lane {0,1} float masks
// hoisted out of the K loop (was: divergent predicated ds_loads with EXEC
// juggling inside the loop). Both used B columns read W2[j][m&1], so one
// shared LDS load feeds both WMMAs. w1/b1 are read as aligned pairs (b64).
// ---------------------------------------------------------------------------

typedef __attribute__((ext_vector_type(2))) float v2f;
typedef __attribute__((ext_vector_type(8))) float v8f;

#define NPTS 524288
#define HID  256

constexpr float DXF       = 20.0f / 524287.0f;    // (X_MAX - X_MIN) / (N-1)
constexpr float DT_F      = 0.01f;
constexpr float SQRT2DT   = 0.14142136f;          // float32(sqrt(2*DT))
constexpr float TWO_LOG2E = 2.8853900817779268f;  // 2/ln(2)

__device__ __forceinline__ float fast_tanh(float z) {
  // tanh(z) = 1 - 2/(exp(2z)+1); v_exp_f32 + v_rcp_f32, saturates correctly.
  float t = __builtin_amdgcn_exp2f(z * TWO_LOG2E);
  return 1.0f - 2.0f * __builtin_amdgcn_rcpf(t + 1.0f);
}

// ---------------------------------------------------------------------------
// Kernel 1: h_psi for all N points + per-block partial sums of
// densities / norm_densities (deterministic: no float atomics).
// Block = 256 threads = 8 waves; each wave owns 16 consecutive points.
// ---------------------------------------------------------------------------
__global__ __launch_bounds__(256) void qpe_hpsi_kernel(
    const float* __restrict__ x,  const float* __restrict__ Vx,
    const float* __restrict__ W1, const float* __restrict__ b1,
    const float* __restrict__ W2, const float* __restrict__ b2,
    float* __restrict__ out, float* __restrict__ partials)
{
  __shared__ __align__(16) float s_w1[HID];
  __shared__ __align__(16) float s_b1[HID];
  __shared__ __align__(16) float s_w2[2 * HID];
  __shared__ float s_stage[8][16][4];  // [wave][point][psi0,psi1,lap0,lap1]
  __shared__ float s_red[2][8];

  const int tid = threadIdx.x;
  for (int i = tid; i < HID; i += 256) { s_w1[i] = W1[i]; s_b1[i] = b1[i]; }
  for (int i = tid; i < 2 * HID; i += 256) { s_w2[i] = W2[i]; }
  __syncthreads();

  const int wave = tid >> 5;
  const int lane = tid & 31;
  const int half = lane >> 4;   // selects K pair {0,1} vs {2,3}
  const int m    = lane & 15;   // M row (for A) / N column (for B)
  const int base = blockIdx.x * 128 + wave * 16;
  const float xm = x[base + m];

  // Branch-free B-element selection, hoisted out of the K loop:
  // column n=m gets  mask_h * W2[j][m&1]            (psi,   n in {0,1})
  //              or  mask_u * w1_j^2 * W2[j][m&1]   (lap,   n in {2,3})
  //              or  0                              (n >= 4)
  const float mask_h = (m < 2) ? 1.0f : 0.0f;
  const float mask_u = (m == 2 || m == 3) ? 1.0f : 0.0f;
  const int   colsel = m & 1;

  v8f acc = {};
  for (int c = 0; c < 64; ++c) {        // 64 chunks of K=4 over HID=256
    const int jb = 4 * c + 2 * half;    // even -> 8B-aligned pair loads
    const v2f w1p = *(const v2f*)&s_w1[jb];
    const v2f b1p = *(const v2f*)&s_b1[jb];
    v2f Ah, Au, Bh, Bu;
#pragma unroll
    for (int v = 0; v < 2; ++v) {
      const float w1j = w1p[v];
      const float h   = fast_tanh(fmaf(xm, w1j, b1p[v]));
      const float u   = -2.0f * h * (1.0f - h * h);
      const float w2v = s_w2[2 * (jb + v) + colsel];  // shared by both columns
      Ah[v] = h;
      Au[v] = u;
      Bh[v] = mask_h * w2v;
      Bu[v] = mask_u * (w1j * w1j * w2v);
    }
    acc = __builtin_amdgcn_wmma_f32_16x16x4_f32(false, Ah, false, Bh,
                                                (short)0, acc, false, false);
    acc = __builtin_amdgcn_wmma_f32_16x16x4_f32(false, Au, false, Bu,
                                                (short)0, acc, false, false);
  }

  // Scatter the 4 useful columns of D into LDS: lane holds column n=m,
  // rows r+8*half for r=0..7.
  if (m < 4) {
#pragma unroll
    for (int r = 0; r < 8; ++r) s_stage[wave][r + 8 * half][m] = acc[r];
  }
  __syncthreads();

  float dens = 0.0f, nrm = 0.0f;
  if (lane < 16) {
    const int p = base + lane;
    const float b20 = b2[0], b21 = b2[1];
    const float psi0 = s_stage[wave][lane][0] + b20;
    const float psi1 = s_stage[wave][lane][1] + b21;
    const float Vp = Vx[p];
    const float hp0 = fmaf(-0.5f, s_stage[wave][lane][2], Vp * psi0); // K_TERM=-0.5
    const float hp1 = fmaf(-0.5f, s_stage[wave][lane][3], Vp * psi1);
    out[2 * p]     = hp0;
    out[2 * p + 1] = hp1;
    dens = psi0 * hp0 + psi1 * hp1;
    nrm  = psi0 * psi0 + psi1 * psi1;
  }
#pragma unroll
  for (int off = 16; off > 0; off >>= 1) {
    dens += __shfl_down(dens, off, 32);
    nrm  += __shfl_down(nrm,  off, 32);
  }
  if (lane == 0) { s_red[0][wave] = dens; s_red[1][wave] = nrm; }
  __syncthreads();
  if (tid == 0) {
    float d = 0.0f, n2 = 0.0f;
#pragma unroll
    for (int w = 0; w < 8; ++w) { d += s_red[0][w]; n2 += s_red[1][w]; }
    partials[2 * blockIdx.x]     = d;
    partials[2 * blockIdx.x + 1] = n2;
  }
}

// ---------------------------------------------------------------------------
// Kernel 2: fixed-order reduction of 4096 block partials -> energy (out[2N])
// and denominator (sumN*DX + 1e-8) into workspace for the scan.
// ---------------------------------------------------------------------------
__global__ __launch_bounds__(256) void qpe_reduce_kernel(
    const float* __restrict__ partials, float* __restrict__ out,
    float* __restrict__ scal)
{
  __shared__ float sd[256];
  __shared__ float sn[256];
  const int tid = threadIdx.x;
  float d = 0.0f, n = 0.0f;
  for (int i = tid; i < 4096; i += 256) {
    d += partials[2 * i];
    n += partials[2 * i + 1];
  }
  sd[tid] = d; sn[tid] = n;
  __syncthreads();
  for (int off = 128; off > 0; off >>= 1) {
    if (tid < off) { sd[tid] += sd[tid + off]; sn[tid] += sn[tid + off]; }
    __syncthreads();
  }
  if (tid == 0) {
    const float sumD = sd[0], sumN = sn[0];
    const float denom = sumN * DXF + 1e-8f;
    out[2 * NPTS] = sumD * DXF / denom;  // energy
    scal[0] = denom;
  }
}

// ---------------------------------------------------------------------------
// Kernel 3: 100-step serial index scan on one wave. log_prob at c+/-1 is
// recomputed on the fly (256-unit contraction split over 32 lanes, full
// shfl_xor butterfly so every lane holds identical sums -> identical,
// deterministic index updates with no broadcast needed).
// ---------------------------------------------------------------------------
__global__ __launch_bounds__(32) void qpe_scan_kernel(
    const float* __restrict__ x,
    const float* __restrict__ W1, const float* __restrict__ b1,
    const float* __restrict__ W2, const float* __restrict__ b2,
    const float* __restrict__ noise, const int* __restrict__ init_idx,
    const float* __restrict__ scal, float* __restrict__ out)
{
  const int lane = threadIdx.x;
  const float denom = scal[0];
  const float b20 = b2[0], b21 = b2[1];
  const float inv2dx = 1.0f / (2.0f * DXF);
  int c = init_idx[0];

  for (int step = 0; step < 100; ++step) {
    float s = 0.0f;  // score == 0 at grid boundaries (matches .at[1:-1])
    if (c > 0 && c < NPTS - 1) {
      float lp0 = 0.0f, lp1 = 0.0f;
#pragma unroll
      for (int side = 0; side < 2; ++side) {
        const int i = side ? (c + 1) : (c - 1);
        const float xi = x[i];
        float p0 = 0.0f, p1 = 0.0f;
        for (int j = lane; j < HID; j += 32) {
          const float h = fast_tanh(fmaf(xi, W1[j], b1[j]));
          p0 = fmaf(h, W2[2 * j],     p0);
          p1 = fmaf(h, W2[2 * j + 1], p1);
        }
#pragma unroll
        for (int off = 16; off > 0; off >>= 1) {
          p0 += __shfl_xor(p0, off, 32);
          p1 += __shfl_xor(p1, off, 32);
        }
        p0 += b20; p1 += b21;
        const float nrm = p0 * p0 + p1 * p1;
        const float lp = logf(nrm / denom + 1e-10f);
        if (side) lp1 = lp; else lp0 = lp;
      }
      s = (lp1 - lp0) * inv2dx;
    }
    const float zn = noise[step];
    const int shift = (int)((s * DT_F + zn * SQRT2DT) / DXF); // trunc toward 0
    c += shift;
    c = c < 0 ? 0 : (c > NPTS - 1 ? NPTS - 1 : c);
  }
  if (lane == 0) out[2 * NPTS + 1] = x[c];
}

// ---------------------------------------------------------------------------
// Host entry. Inputs (setup_inputs order):
//   0:x[N] 1:V_x[N] 2:W1[256] 3:b1[256] 4:W2[512] 5:b2[2] 6:noise[100] 7:init_idx[1]
// Output: h_psi[N*2] ++ energy[1] ++ x_collapsed[1]  (2N+2 floats)
// Workspace: 4096*2 partial floats + 1 scalar (~33 KB).
// ---------------------------------------------------------------------------
extern "C" void kernel_launch(void* const* d_in, const int* in_sizes, int n_in,
                              void* d_out, int out_size, void* d_ws, size_t ws_size,
                              hipStream_t stream) {
  const float* x        = (const float*)d_in[0];
  const float* Vx       = (const float*)d_in[1];
  const float* W1       = (const float*)d_in[2];
  const float* b1       = (const float*)d_in[3];
  const float* W2       = (const float*)d_in[4];
  const float* b2       = (const float*)d_in[5];
  const float* noise    = (const float*)d_in[6];
  const int*   init_idx = (const int*)d_in[7];
  float* out = (float*)d_out;

  float* partials = (float*)d_ws;          // 4096 * 2 floats
  float* scal     = partials + 4096 * 2;   // 1 float: denominator

  qpe_hpsi_kernel<<<NPTS / 128, 256, 0, stream>>>(x, Vx, W1, b1, W2, b2, out, partials);
  qpe_reduce_kernel<<<1, 256, 0, stream>>>(partials, out, scal);
  qpe_scan_kernel<<<1, 32, 0, stream>>>(x, W1, b1, W2, b2, noise, init_idx, scal, out);
}